// SupCRLoss_74912819577580
// MI455X (gfx1250) — compile-verified
//
#include <hip/hip_runtime.h>
#include <hip/hip_bf16.h>

// CDNA5 (gfx1250) wave32 WMMA types
typedef float v2f __attribute__((ext_vector_type(2)));
typedef float v8f __attribute__((ext_vector_type(8)));

#define NMAX 2048
#define INV_T 14.285714285714286f   // 1 / 0.07

// ---------------------------------------------------------------------------
// Block-wide sum reduction for blockDim.x == 256
// ---------------------------------------------------------------------------
__device__ __forceinline__ float block_reduce256(float v) {
    __shared__ float s[256];
    int t = threadIdx.x;
    s[t] = v;
    __syncthreads();
#pragma unroll
    for (int o = 128; o > 0; o >>= 1) {
        if (t < o) s[t] += s[t + o];
        __syncthreads();
    }
    float r = s[0];
    __syncthreads();
    return r;
}

// ---------------------------------------------------------------------------
// 1) Per-row squared norms of cf, lab vector. cf row r = features row
//    (r % B)*2 + r/B (V == 2), contiguous 128 floats. One wave per row.
// ---------------------------------------------------------------------------
__global__ void prep_kernel(const float* __restrict__ feat,
                            const float* __restrict__ labels,
                            float* __restrict__ sq, float* __restrict__ lab,
                            int Bsz, int N, int D) {
    int wave = threadIdx.x >> 5;
    int lane = threadIdx.x & 31;
    int r = blockIdx.x * (blockDim.x >> 5) + wave;
    if (r >= N) return;
    int fr = (r % Bsz) * 2 + (r / Bsz);
    const float4* row = (const float4*)(feat + (size_t)fr * D);
    float acc = 0.f;
    for (int k4 = lane; k4 < (D >> 2); k4 += 32) {
        float4 v = row[k4];
        acc += v.x * v.x + v.y * v.y + v.z * v.z + v.w * v.w;
    }
#pragma unroll
    for (int o = 16; o > 0; o >>= 1) acc += __shfl_xor(acc, o, 32);
    if (lane == 0) {
        sq[r]  = acc;
        lab[r] = labels[r % Bsz];
    }
}

// ---------------------------------------------------------------------------
// 2) Gram tile via V_WMMA_F32_16X16X4_F32 (wave32, one 16x16 tile per wave).
//    Produces logits0 (diag zeroed) and exp_logits directly.
//    A frag (16x4 f32, 2 VGPR): lane<16 -> M=lane,K={0,1}; lane>=16 -> K={2,3}
//    B frag (4x16 f32, 2 VGPR): lane<16 -> N=lane,K={0,1}; lane>=16 -> K={2,3}
//    C/D (16x16 f32, 8 VGPR):   vgpr v -> M = v + 8*(lane>=16), N = lane&15
// ---------------------------------------------------------------------------
__global__ void gram_kernel(const float* __restrict__ feat,
                            const float* __restrict__ sq,
                            float* __restrict__ logits0,
                            float* __restrict__ elog,
                            int Bsz, int N) {
    const int D = 128;
    int lane = threadIdx.x;          // blockDim.x == 32
    int half = lane >> 4;            // 0 or 1
    int mr   = lane & 15;
    int i0 = blockIdx.y * 16;
    int j0 = blockIdx.x * 16;
    if (i0 >= N || j0 >= N) return;

    int ra = i0 + mr;                // A row for this lane
    int rb = j0 + mr;                // B column (= cf row) for this lane
    int fra = (ra % Bsz) * 2 + (ra / Bsz);
    int frb = (rb % Bsz) * 2 + (rb / Bsz);
    const float* ap = feat + (size_t)fra * D + 2 * half;
    const float* bp = feat + (size_t)frb * D + 2 * half;

    v8f acc = {};
#pragma unroll
    for (int k = 0; k < D; k += 4) {
        v2f a = *(const v2f*)(ap + k);
        v2f b = *(const v2f*)(bp + k);
        // D = A x B + C  (f32 matrix pipe), emits v_wmma_f32_16x16x4_f32
        acc = __builtin_amdgcn_wmma_f32_16x16x4_f32(
            /*neg_a=*/false, a, /*neg_b=*/false, b,
            /*c_mod=*/(short)0, acc, /*reuse_a=*/false, /*reuse_b=*/false);
    }

    int   j   = j0 + mr;
    float sqj = sq[j];
#pragma unroll
    for (int v = 0; v < 8; ++v) {
        int i = i0 + v + 8 * half;
        float dist = sq[i] + sqj - 2.0f * acc[v];
        float lg = __expf(-dist) * INV_T;
        float el = __expf(lg);
        size_t idx = (size_t)i * N + j;
        logits0[idx] = (i == j) ? 0.0f : lg;
        elog[idx]    = el;
    }
}

// ---------------------------------------------------------------------------
// 3) rowA[i] = sum_j logits0[i][j]
// ---------------------------------------------------------------------------
__global__ void rowsum_kernel(const float* __restrict__ logits0,
                              float* __restrict__ rowA, int N) {
    int i = blockIdx.x;
    float acc = 0.f;
    for (int j = threadIdx.x; j < N; j += blockDim.x)
        acc += logits0[(size_t)i * N + j];
    float tot = block_reduce256(acc);
    if (threadIdx.x == 0) rowA[i] = tot;
}

// ---------------------------------------------------------------------------
// 4) O(N^3) masked-sum pass.
//    negative[i][m] = sum_j elog[i][j] * (|lab_j - lab_m| >= |lab_i - lab_m|)
//    Stage elog row i (8KB) and lab (8KB) in LDS; one thread per column m;
//    deterministic per-block partial sums of log(negative) (no atomics).
// ---------------------------------------------------------------------------
__global__ void negative_kernel(const float* __restrict__ elog,
                                const float* __restrict__ lab,
                                float* __restrict__ rowBpart,
                                int N, int MB) {
    __shared__ float s_e[NMAX];
    __shared__ float s_l[NMAX];
    int i = blockIdx.y;
    int m = blockIdx.x * blockDim.x + threadIdx.x;
    for (int j = threadIdx.x; j < N; j += blockDim.x) {
        s_e[j] = elog[(size_t)i * N + j];
        s_l[j] = lab[j];
    }
    __syncthreads();

    float v = 0.0f;
    if (m < N) {
        float labm = s_l[m];
        float thr  = fabsf(s_l[i] - labm);
        float acc  = 0.f;
#pragma unroll 4
        for (int j = 0; j < N; ++j) {
            float d = fabsf(s_l[j] - labm);
            acc += (d >= thr) ? s_e[j] : 0.0f;
        }
        v = (m == i) ? 0.0f : __logf(acc);   // negative[i][i] := 1 -> log = 0
    }
    float tot = block_reduce256(v);
    if (threadIdx.x == 0) rowBpart[(size_t)i * MB + blockIdx.x] = tot;
}

// ---------------------------------------------------------------------------
// 5) loss = -(T/BT) * mean_i( (rowA[i] - rowBsum[i]) / (N-1) ), T/BT = 1
// ---------------------------------------------------------------------------
__global__ void finalize_kernel(const float* __restrict__ rowA,
                                const float* __restrict__ rowBpart,
                                float* __restrict__ out, int N, int MB) {
    float acc = 0.f;
    for (int idx = threadIdx.x; idx < N; idx += blockDim.x) acc += rowA[idx];
    for (int idx = threadIdx.x; idx < N * MB; idx += blockDim.x)
        acc -= rowBpart[idx];
    float tot = block_reduce256(acc);
    if (threadIdx.x == 0)
        out[0] = -tot / ((float)N * (float)(N - 1));
}

// ---------------------------------------------------------------------------
extern "C" void kernel_launch(void* const* d_in, const int* in_sizes, int n_in,
                              void* d_out, int out_size, void* d_ws, size_t ws_size,
                              hipStream_t stream) {
    const float* feat   = (const float*)d_in[0];   // [B, 2, 128] f32
    const float* labels = (const float*)d_in[1];   // [B] f32
    const int Bsz = in_sizes[1];
    const int N   = 2 * Bsz;                       // V == 2
    const int D   = 128;
    const int MB  = (N + 255) / 256;

    // Workspace layout (floats): sq[N] | lab[N] | rowA[N] | rowBpart[N*MB] |
    //                            logits0[N*N] | elog[N*N]   (~33.7 MB)
    float* ws       = (float*)d_ws;
    float* sq       = ws;
    float* lab      = ws + N;
    float* rowA     = ws + 2 * (size_t)N;
    float* rowBpart = ws + 3 * (size_t)N;
    float* logits0  = rowBpart + (size_t)N * MB;
    float* elog     = logits0 + (size_t)N * N;

    prep_kernel<<<(N + 7) / 8, 256, 0, stream>>>(feat, labels, sq, lab, Bsz, N, D);

    dim3 ggrid((N + 15) / 16, (N + 15) / 16);
    gram_kernel<<<ggrid, 32, 0, stream>>>(feat, sq, logits0, elog, Bsz, N);

    rowsum_kernel<<<N, 256, 0, stream>>>(logits0, rowA, N);

    dim3 ngrid(MB, N);
    negative_kernel<<<ngrid, 256, 0, stream>>>(elog, lab, rowBpart, N, MB);

    finalize_kernel<<<1, 256, 0, stream>>>(rowA, rowBpart, (float*)d_out, N, MB);
}